// Shift_14422500180434
// MI455X (gfx1250) — compile-verified
//
#include <hip/hip_runtime.h>

// Shift op: out[n, c*9 + (dy*3+dx), y, x] = x[n, c, y+dy-1, x+dx-1] (zero-padded)
// x: (32, 64, 56, 56) f32  ->  out: (32, 576, 56, 56) f32
//
// One workgroup per (n,c) plane. TDM (iterate mode) stages the 56x56 plane
// into the interior of a zero-haloed 58x58 LDS tile (row pitch 58); the DMA
// overlaps with threads zeroing the halo (disjoint addresses). Fan-out needs
// no bounds checks: per 4-wide chunk, read a 3x6 LDS window and emit 9 NT
// 128-bit stores at immediate offsets off one base address.

typedef __attribute__((ext_vector_type(4))) unsigned int uint32x4;
typedef __attribute__((ext_vector_type(8))) int          int32x8;
typedef __attribute__((ext_vector_type(4))) int          int32x4;
typedef __attribute__((ext_vector_type(4))) float        v4f;

#define H 56
#define W 56
#define PW 58                      // padded row pitch (floats)
#define PLANE (H * W)              // 3136 floats per plane
#define NCH (H * (W / 4))          // 784 float4 chunks per plane

__global__ __launch_bounds__(256) void shift_tdm_kernel(
    const float* __restrict__ x, float* __restrict__ out)
{
    __shared__ float padt[PW * PW];            // 58x58 = 13456 bytes

    const unsigned plane = blockIdx.x;         // 0 .. 2047, uniform (SGPR)
    const int tid = (int)threadIdx.x;

    // ---- Kick off TDM first: 56 rows of 56 floats, LDS pitch 58 ----
    if (tid < 32) {
        unsigned long long gaddr =
            (unsigned long long)(const void*)x + (unsigned long long)plane * (PLANE * 4ull);
        unsigned lds_addr =
            (unsigned)(unsigned long long)(const void*)&padt[1 * PW + 1]; // interior origin

        // D# group 0: count=1 (valid), lds_addr, global_addr[56:0], type=2
        uint32x4 g0;
        g0[0] = 1u;
        g0[1] = lds_addr;
        g0[2] = (unsigned)(gaddr & 0xFFFFFFFFull);
        g0[3] = (unsigned)((gaddr >> 32) & 0x01FFFFFFull) | (2u << 30);

        // D# group 1: data_size=4B, iterate_enable, 56-wide tensor, 56x1 tile
        int32x8 g1;
        g1[0] = (int)((2u << 16) | (1u << 19)); // data_size=2 (4B), iterate_enable=1
        g1[1] = (int)(56u << 16);               // [63:48]   tensor_dim0 = 56
        g1[2] = (int)(56u << 16);               // [95:80]   tensor_dim1 = 56
        g1[3] = (int)(56u << 16);               // [127:112] tile_dim0   = 56
        g1[4] = (int)(1u);                      // [143:128] tile_dim1 = 1; tile_dim2 = 0
        g1[5] = (int)(56u);                     // [191:160] tensor_dim0_stride = 56
        g1[6] = (int)(3136u << 16);             // [223:208] tensor_dim1_stride = 3136
        g1[7] = 0;

        // D# group 2 (iterate mode): lds+global increments, 56 iterations
        int32x4 g2;
        g2[0] = 0;                              // tensor_dim2 = 0
        g2[1] = (int)PW;                        // lds_addr_increment    = 58 elements
        g2[2] = (int)W;                         // global_addr_increment = 56 elements
        g2[3] = (int)(55u << 16);               // iterate_count = 55 -> 56 iterations

        int32x4 g3 = {0, 0, 0, 0};
        int32x8 g4 = {0, 0, 0, 0, 0, 0, 0, 0};  // VADDR4 slot: unused

        __builtin_amdgcn_tensor_load_to_lds(g0, g1, g2, g3, g4, 0);
    }

    // ---- Zero the halo ring while the DMA runs (disjoint addresses) ----
    if (tid < 228) {
        int r, c;
        if      (tid < 58)  { r = 0;            c = tid;       }
        else if (tid < 116) { r = PW - 1;       c = tid - 58;  }
        else if (tid < 172) { r = tid - 115;    c = 0;         }   // rows 1..56
        else                { r = tid - 171;    c = PW - 1;    }   // rows 1..56
        padt[r * PW + c] = 0.0f;
    }

    // ---- Wave 0 holds the barrier until the DMA lands ----
    if (tid < 32) __builtin_amdgcn_s_wait_tensorcnt(0);
    __syncthreads();                            // publish padded tile to all 8 waves

    // ---- Fan out: no bounds checks; 9 NT b128 stores per chunk ----
    float* outbase = out + (unsigned long long)plane * 9ull * PLANE;

    for (int cc = tid; cc < NCH; cc += 256) {
        const int y  = cc / 14;                 // output row 0..55
        const int x0 = (cc % 14) * 4;           // output col 0,4,..,52

        // 3x6 padded window: f[r][j] = xp[y+r][x0+j]
        float f[3][6];
        #pragma unroll
        for (int r = 0; r < 3; ++r) {
            const float* row = &padt[(y + r) * PW + x0];
            #pragma unroll
            for (int j = 0; j < 6; ++j) f[r][j] = row[j];
        }

        float* dst0 = outbase + y * W + x0;
        #pragma unroll
        for (int s = 0; s < 9; ++s) {
            const int dy = s / 3, dx = s % 3;
            v4f v;
            v[0] = f[dy][dx + 0];
            v[1] = f[dy][dx + 1];
            v[2] = f[dy][dx + 2];
            v[3] = f[dy][dx + 3];
            __builtin_nontemporal_store(v, (v4f*)(dst0 + s * PLANE));
        }
    }
}

extern "C" void kernel_launch(void* const* d_in, const int* in_sizes, int n_in,
                              void* d_out, int out_size, void* d_ws, size_t ws_size,
                              hipStream_t stream)
{
    const float* x = (const float*)d_in[0];
    float* out = (float*)d_out;

    dim3 grid(32 * 64);   // one block per (n, c) plane
    dim3 block(256);      // 8 wave32 waves
    shift_tdm_kernel<<<grid, block, 0, stream>>>(x, out);
}